// RepulsionLoss_1580547972159
// MI455X (gfx1250) — compile-verified
//
#include <hip/hip_runtime.h>
#include <math.h>

typedef float v2f __attribute__((ext_vector_type(2)));
typedef float v8f __attribute__((ext_vector_type(8)));

#define NPTS     4096
#define NBATCH   8
#define INV_H2   (1.0f / (0.03f * 0.03f))
#define BIGF     3.0e38f

#define WPB      4                    // waves per block
#define TPB      (WPB * 32)           // 128 threads
#define QPW      32                   // queries per wave (two 16-row A tiles)
#define NWAVES   (NBATCH * NPTS / QPW)   // 1024
#define NBLOCKS  (NWAVES / WPB)          // 256

static __device__ __forceinline__ void insert4(float d, float& t0, float& t1,
                                               float& t2, float& t3) {
    float x = d, m;
    m = fminf(t0, x); x = fmaxf(t0, x); t0 = m;
    m = fminf(t1, x); x = fmaxf(t1, x); t1 = m;
    m = fminf(t2, x); x = fmaxf(t2, x); t2 = m;
    t3 = fminf(t3, x);
}

__global__ __launch_bounds__(TPB)
void repulsion_partial_kernel(const float* __restrict__ pc,
                              float* __restrict__ partial) {
    // 32x16 f32 distance staging tile per wave
    __shared__ float lds[WPB * 32 * 16];

    const int tid  = threadIdx.x;
    const int lane = tid & 31;
    const int wid  = tid >> 5;
    const int w    = blockIdx.x * WPB + wid;     // global wave id
    const int qbase = w * QPW;                   // global query base
    const int b    = qbase >> 12;                // batch (4096 pts/batch)
    const int q0   = qbase & (NPTS - 1);         // query base within batch

    const float* __restrict__ P = pc + (size_t)b * NPTS * 3;
    float* __restrict__ wls = &lds[wid * 32 * 16];

    const int  lm = lane & 15;
    const bool hi = lane >= 16;

    // ---- A tiles (16x4 f32, K=4):  row m = (-2x, -2y, -2z, 1)
    // VGPR0: lanes0-15 K=0, lanes16-31 K=2 ; VGPR1: lanes0-15 K=1, lanes16-31 K=3
    v2f a0, a1;
    {
        int qi = q0 + lm;
        float x = P[qi * 3 + 0], y = P[qi * 3 + 1], z = P[qi * 3 + 2];
        a0.x = hi ? (-2.0f * z) : (-2.0f * x);
        a0.y = hi ? (1.0f)      : (-2.0f * y);
        qi = q0 + 16 + lm;
        x = P[qi * 3 + 0]; y = P[qi * 3 + 1]; z = P[qi * 3 + 2];
        a1.x = hi ? (-2.0f * z) : (-2.0f * x);
        a1.y = hi ? (1.0f)      : (-2.0f * y);
    }

    // ---- this lane's own query point
    const int qown = q0 + lane;
    float sqo;
    {
        float x = P[qown * 3 + 0], y = P[qown * 3 + 1], z = P[qown * 3 + 2];
        sqo = x * x + y * y + z * z;
    }

    // running 4 smallest non-self distances for this lane's query
    float t0 = BIGF, t1 = BIGF, t2 = BIGF, t3 = BIGF;

    for (int nt = 0; nt < NPTS / 16; ++nt) {
        const int c0 = nt * 16;

        // ---- B tile (4x16 f32): col n = (x, y, z, ||q_n||^2)
        const int ci = c0 + lm;
        float cx = P[ci * 3 + 0], cy = P[ci * 3 + 1], cz = P[ci * 3 + 2];
        float sqc = cx * cx + cy * cy + cz * cz;
        v2f bt;
        bt.x = hi ? cz  : cx;   // K=2 : K=0
        bt.y = hi ? sqc : cy;   // K=3 : K=1

        v8f czero = {};
        // C[m][n] = ||q_n||^2 - 2 <p_m, q_n>
        v8f g0 = __builtin_amdgcn_wmma_f32_16x16x4_f32(
            false, a0, false, bt, (short)0, czero, false, false);
        v8f g1 = __builtin_amdgcn_wmma_f32_16x16x4_f32(
            false, a1, false, bt, (short)0, czero, false, false);

        // ---- scatter C tiles to LDS as [m (0..31)][n (0..15)]
        // C layout: VGPR r, lanes0-15 -> M=r, lanes16-31 -> M=8+r, N=lane&15
        {
            const int mbase = hi ? 8 : 0;
            float* sp = &wls[mbase * 16 + lm];
#pragma unroll
            for (int r = 0; r < 8; ++r) sp[r * 16] = g0[r];
#pragma unroll
            for (int r = 0; r < 8; ++r) sp[(16 + r) * 16] = g1[r];
        }

        // same-wave DS ops are in-order; just drain them before the row reads
        asm volatile("s_wait_dscnt 0" ::: "memory");

        // ---- each lane reads its own query's 16 distances
        const float4* rp = (const float4*)&wls[lane * 16];
        float4 r0 = rp[0], r1 = rp[1], r2 = rp[2], r3 = rp[3];
        float gv[16] = {r0.x, r0.y, r0.z, r0.w, r1.x, r1.y, r1.z, r1.w,
                        r2.x, r2.y, r2.z, r2.w, r3.x, r3.y, r3.z, r3.w};
#pragma unroll
        for (int n = 0; n < 16; ++n) {
            float d = sqo + gv[n];                 // ||p||^2 + ||q||^2 - 2<p,q>
            d = fmaxf(d, 0.0f);
            d = ((c0 + n) == qown) ? BIGF : d;     // skip self-match
            insert4(d, t0, t1, t2, t3);
        }
    }

    // ---- loss contribution: sum_k -d_k * exp(-d_k / H^2)
    float acc = -(t0 * __expf(-t0 * INV_H2) + t1 * __expf(-t1 * INV_H2) +
                  t2 * __expf(-t2 * INV_H2) + t3 * __expf(-t3 * INV_H2));

    // wave reduction (wave32), fixed order -> deterministic
#pragma unroll
    for (int off = 16; off > 0; off >>= 1)
        acc += __shfl_xor(acc, off, 32);

    if (lane == 0) partial[w] = acc;
}

__global__ __launch_bounds__(256)
void repulsion_reduce_kernel(const float* __restrict__ partial,
                             float* __restrict__ out) {
    __shared__ float s[256];
    const int t = threadIdx.x;
    // fixed summation order -> deterministic
    float v = partial[t] + partial[t + 256] + partial[t + 512] + partial[t + 768];
    s[t] = v;
    __syncthreads();
#pragma unroll
    for (int off = 128; off > 0; off >>= 1) {
        if (t < off) s[t] += s[t + off];
        __syncthreads();
    }
    if (t == 0) out[0] = s[0];
}

extern "C" void kernel_launch(void* const* d_in, const int* in_sizes, int n_in,
                              void* d_out, int out_size, void* d_ws, size_t ws_size,
                              hipStream_t stream) {
    (void)in_sizes; (void)n_in; (void)out_size; (void)ws_size;
    const float* pc = (const float*)d_in[0];      // (8, 4096, 3) f32
    float* out      = (float*)d_out;              // scalar f32
    float* partials = (float*)d_ws;               // NWAVES floats of scratch

    repulsion_partial_kernel<<<NBLOCKS, TPB, 0, stream>>>(pc, partials);
    repulsion_reduce_kernel<<<1, 256, 0, stream>>>(partials, out);
}